// Model_62483184222606
// MI455X (gfx1250) — compile-verified
//
#include <hip/hip_runtime.h>
#include <hip/hip_bf16.h>

// ---------------------------------------------------------------------------
// Hetero-GraphSAGE link prediction for MI455X (gfx1250), fp32 WMMA path.
// Memory-bound workload (edge scatter/gather ~4.6GB >> 70 GFLOP), so we keep
// full fp32 precision and use V_WMMA_F32_16X16X4_F32 for all dense GEMMs.
// ---------------------------------------------------------------------------

#define NUc 100000
#define NMc 50000
#define Hc  128
#define Ec  600000
#define ELc 400000
#define Lc  3

typedef __attribute__((ext_vector_type(2))) float v2f;
typedef __attribute__((ext_vector_type(8))) float v8f;

__device__ __forceinline__ v8f wmma_f32(v2f a, v2f b, v8f c) {
    // 8 args: (neg_a, A, neg_b, B, c_mod, C, reuse_a, reuse_b)
    return __builtin_amdgcn_wmma_f32_16x16x4_f32(
        false, a, false, b, (short)0, c, false, false);
}

// ---------------------------------------------------------------------------
// xu0[i,:] = user_emb[user_n_id[i],:]
// ---------------------------------------------------------------------------
__global__ __launch_bounds__(256) void gather_kernel(
    float* __restrict__ dst, const int* __restrict__ ids,
    const float* __restrict__ emb, int n)
{
    size_t g = (size_t)blockIdx.x * blockDim.x + threadIdx.x;
    size_t tot = (size_t)n * Hc;
    if (g >= tot) return;
    int i = (int)(g >> 7);
    int h = (int)(g & 127);
    dst[g] = emb[(size_t)ids[i] * Hc + h];
}

// ---------------------------------------------------------------------------
// Scatter-add: msum[dst[e],:] += x[src[e],:]   (32 lanes per edge, float4)
// msum has row stride 128.
// ---------------------------------------------------------------------------
__global__ __launch_bounds__(256) void scatter_add_kernel(
    float* __restrict__ msum, const float* __restrict__ x, int ldx,
    const int* __restrict__ src, const int* __restrict__ dst, int nE)
{
    int g = blockIdx.x * blockDim.x + threadIdx.x;
    int e = g >> 5;
    int lane = g & 31;
    if (e >= nE) return;
    int s = src[e];
    int d = dst[e];
    const float4 v = ((const float4*)(x + (size_t)s * ldx))[lane];
    float* mp = msum + (size_t)d * Hc + lane * 4;
    atomicAdd(mp + 0, v.x);
    atomicAdd(mp + 1, v.y);
    atomicAdd(mp + 2, v.z);
    atomicAdd(mp + 3, v.w);
}

__global__ __launch_bounds__(256) void count_kernel(
    float* __restrict__ cnt, const int* __restrict__ dst, int nE)
{
    int e = blockIdx.x * blockDim.x + threadIdx.x;
    if (e >= nE) return;
    atomicAdd(&cnt[dst[e]], 1.0f);
}

// m[i,h] /= max(cnt[i], 1)
__global__ __launch_bounds__(256) void normalize_kernel(
    float* __restrict__ m, const float* __restrict__ cnt, int n)
{
    size_t g = (size_t)blockIdx.x * blockDim.x + threadIdx.x;
    if (g >= (size_t)n * Hc) return;
    int i = (int)(g >> 7);
    float c = fmaxf(cnt[i], 1.0f);
    m[g] *= (1.0f / c);
}

// ---------------------------------------------------------------------------
// Generic WMMA GEMM, N = 128 output columns:
//   out[row, :] = act( A0[row,0:k0] @ W0  +  A1[row,0:k1] @ W1  + bias )
// A tile (16 x (k0+k1)) staged in LDS; block = 256 threads = 8 waves, each
// wave owns a 16x16 N-tile. WMMA fragment layout per ISA 7.12.2:
//   A (16x4 f32):  lanes 0-15 -> M=lane, {v0,v1}={K,K+1};
//                  lanes 16-31 -> M=lane-16, {v0,v1}={K+2,K+3}
//   B (4x16 f32):  lanes 0-15 -> N=lane, {v0,v1}=rows{K,K+1};
//                  lanes 16-31 -> N=lane-16, rows{K+2,K+3}
//   C (16x16 f32): VGPR r -> M = r + 8*(lane>=16), N = lane&15
// ---------------------------------------------------------------------------
#define AS_STRIDE 385   // 385 % 64 == 1 -> conflict-free column reads

__global__ __launch_bounds__(256) void wmma_gemm128(
    float* __restrict__ out, int ldOut,
    const float* __restrict__ A0, int ldA0, int k0,
    const float* __restrict__ A1, int ldA1, int k1,
    const float* __restrict__ W0, const float* __restrict__ W1,
    const float* __restrict__ bias, int doRelu)
{
    __shared__ float As[16 * AS_STRIDE];
    const int tid  = threadIdx.x;
    const int row0 = blockIdx.x * 16;
    const int kTot = k0 + k1;

    for (int r = 0; r < 16; ++r) {
        const float* a0r = A0 + (size_t)(row0 + r) * ldA0;
        const float* a1r = (k1 > 0) ? (A1 + (size_t)(row0 + r) * ldA1) : A0;
        for (int c = tid; c < kTot; c += 256)
            As[r * AS_STRIDE + c] = (c < k0) ? a0r[c] : a1r[c - k0];
    }
    __syncthreads();

    const int wave = tid >> 5;
    const int lane = tid & 31;
    const int hi   = lane >> 4;   // 0 or 1
    const int lo   = lane & 15;
    const int n0   = wave * 16;

    v8f c = {};
    for (int k = 0; k < kTot; k += 4) {
        v2f a, b;
        const int ak = k + 2 * hi;
        a.x = As[lo * AS_STRIDE + ak];
        a.y = As[lo * AS_STRIDE + ak + 1];
        const float* Wp = (k < k0) ? (W0 + (size_t)k * 128)
                                   : (W1 + (size_t)(k - k0) * 128);
        b.x = Wp[(2 * hi) * 128 + n0 + lo];
        b.y = Wp[(2 * hi + 1) * 128 + n0 + lo];
        c = wmma_f32(a, b, c);
    }

    const float bv = bias[n0 + lo];
    #pragma unroll
    for (int r = 0; r < 8; ++r) {
        float v = c[r] + bv;
        if (doRelu) v = fmaxf(v, 0.0f);
        out[(size_t)(row0 + r + 8 * hi) * ldOut + n0 + lo] = v;
    }
}

// ---------------------------------------------------------------------------
// Fused classifier: 16 edges per block.
//   comb = [xu_proj[u], xm_proj[m]]           (16 x 256, LDS)
//   h1 = relu(bn1(comb @ W1 + b1))            (16 x 128, LDS, 8 waves WMMA)
//   h2 = relu(bn2(h1 @ W2 + b2))              (16 x 64,  LDS, 4 waves WMMA)
//   out = h2 @ W3 + b3                        (16 x 1)
// ---------------------------------------------------------------------------
__global__ __launch_bounds__(256) void classifier_kernel(
    const float* __restrict__ xu_proj, const float* __restrict__ xm_proj,
    const int* __restrict__ eu, const int* __restrict__ em,
    const float* __restrict__ W1, const float* __restrict__ b1,
    const float* __restrict__ g1, const float* __restrict__ be1,
    const float* __restrict__ W2, const float* __restrict__ b2,
    const float* __restrict__ g2, const float* __restrict__ be2,
    const float* __restrict__ W3, const float* __restrict__ b3,
    float* __restrict__ out)
{
    __shared__ float comb[16][257];
    __shared__ float h1[16][129];
    __shared__ float h2[16][65];

    const int e0  = blockIdx.x * 16;
    const int tid = threadIdx.x;

    for (int r = 0; r < 16; ++r) {
        const int u = eu[e0 + r];
        const int m = em[e0 + r];
        const int cdx = tid;   // 256 threads == 256 columns
        comb[r][cdx] = (cdx < 128) ? xu_proj[(size_t)u * Hc + cdx]
                                   : xm_proj[(size_t)m * Hc + (cdx - 128)];
    }
    __syncthreads();

    const int wave = tid >> 5;
    const int lane = tid & 31;
    const int hi   = lane >> 4;
    const int lo   = lane & 15;
    const float inv = rsqrtf(1.0f + 1e-5f);

    // --- GEMM1: K=256, N=128 (all 8 waves) ---
    {
        const int n0 = wave * 16;
        v8f c = {};
        #pragma unroll 8
        for (int k = 0; k < 256; k += 4) {
            v2f a, b;
            a.x = comb[lo][k + 2 * hi];
            a.y = comb[lo][k + 2 * hi + 1];
            b.x = W1[(size_t)(k + 2 * hi) * 128 + n0 + lo];
            b.y = W1[(size_t)(k + 2 * hi + 1) * 128 + n0 + lo];
            c = wmma_f32(a, b, c);
        }
        const float s  = g1[n0 + lo] * inv;
        const float bb = b1[n0 + lo];
        const float bt = be1[n0 + lo];
        #pragma unroll
        for (int r = 0; r < 8; ++r) {
            float v = (c[r] + bb) * s + bt;
            h1[r + 8 * hi][n0 + lo] = fmaxf(v, 0.0f);
        }
    }
    __syncthreads();

    // --- GEMM2: K=128, N=64 (waves 0..3) ---
    if (wave < 4) {
        const int n0 = wave * 16;
        v8f c = {};
        #pragma unroll 8
        for (int k = 0; k < 128; k += 4) {
            v2f a, b;
            a.x = h1[lo][k + 2 * hi];
            a.y = h1[lo][k + 2 * hi + 1];
            b.x = W2[(size_t)(k + 2 * hi) * 64 + n0 + lo];
            b.y = W2[(size_t)(k + 2 * hi + 1) * 64 + n0 + lo];
            c = wmma_f32(a, b, c);
        }
        const float s  = g2[n0 + lo] * inv;
        const float bb = b2[n0 + lo];
        const float bt = be2[n0 + lo];
        #pragma unroll
        for (int r = 0; r < 8; ++r) {
            float v = (c[r] + bb) * s + bt;
            h2[r + 8 * hi][n0 + lo] = fmaxf(v, 0.0f);
        }
    }
    __syncthreads();

    // --- final: out = h2 @ W3 + b3 ---
    if (tid < 16) {
        float acc = b3[0];
        #pragma unroll 16
        for (int k = 0; k < 64; ++k) acc += h2[tid][k] * W3[k];
        out[e0 + tid] = acc;
    }
}

// ---------------------------------------------------------------------------
// Host-side orchestration (graph-capture safe: only kernel launches and
// hipMemsetAsync on `stream`).
// ---------------------------------------------------------------------------
extern "C" void kernel_launch(void* const* d_in, const int* in_sizes, int n_in,
                              void* d_out, int out_size, void* d_ws, size_t ws_size,
                              hipStream_t stream)
{
    (void)in_sizes; (void)n_in; (void)out_size; (void)ws_size;

    const int*   user_n_id = (const int*)  d_in[0];
    const float* movie_x   = (const float*)d_in[1];
    const int*   ei_rates  = (const int*)  d_in[2];   // [2,E]
    const int*   ei_rev    = (const int*)  d_in[3];   // [2,E]
    const int*   eli       = (const int*)  d_in[4];   // [2,EL]
    const float* user_emb  = (const float*)d_in[5];
    const float* Wl_rates  = (const float*)d_in[6];
    const float* Wr_rates  = (const float*)d_in[7];
    const float* b_rates   = (const float*)d_in[8];
    const float* Wl_rev    = (const float*)d_in[9];
    const float* Wr_rev    = (const float*)d_in[10];
    const float* b_rev     = (const float*)d_in[11];
    const float* Wp_user   = (const float*)d_in[12];
    const float* bp_user   = (const float*)d_in[13];
    const float* Wp_movie  = (const float*)d_in[14];
    const float* bp_movie  = (const float*)d_in[15];
    const float* W1  = (const float*)d_in[16];
    const float* b1  = (const float*)d_in[17];
    const float* g1  = (const float*)d_in[18];
    const float* be1 = (const float*)d_in[19];
    const float* W2  = (const float*)d_in[20];
    const float* b2  = (const float*)d_in[21];
    const float* g2  = (const float*)d_in[22];
    const float* be2 = (const float*)d_in[23];
    const float* W3  = (const float*)d_in[24];
    const float* b3  = (const float*)d_in[25];
    float* out = (float*)d_out;

    float* ws = (float*)d_ws;
    size_t off = 0;
    float* xu0     = ws + off; off += (size_t)NUc * Hc;
    float* xu_cat  = ws + off; off += (size_t)NUc * 3 * Hc;
    float* xm_cat  = ws + off; off += (size_t)NMc * 3 * Hc;
    float* m_u     = ws + off; off += (size_t)NUc * Hc;
    float* m_m     = ws + off; off += (size_t)NMc * Hc;
    float* cnt_u   = ws + off; off += NUc;
    float* cnt_m   = ws + off; off += NMc;
    float* xu_proj = ws + off; off += (size_t)NUc * Hc;
    float* xm_proj = ws + off; off += (size_t)NMc * Hc;

    // 1) gather user embeddings
    {
        size_t n = (size_t)NUc * Hc;
        gather_kernel<<<(unsigned)((n + 255) / 256), 256, 0, stream>>>(
            xu0, user_n_id, user_emb, NUc);
    }

    const float* xs_u = xu0;     int ld_u = Hc;
    const float* xs_m = movie_x; int ld_m = Hc;

    // 2) SAGE layers
    for (int l = 0; l < Lc; ++l) {
        hipMemsetAsync(m_m,   0, (size_t)NMc * Hc * sizeof(float), stream);
        hipMemsetAsync(m_u,   0, (size_t)NUc * Hc * sizeof(float), stream);
        hipMemsetAsync(cnt_m, 0, (size_t)NMc * sizeof(float), stream);
        hipMemsetAsync(cnt_u, 0, (size_t)NUc * sizeof(float), stream);

        scatter_add_kernel<<<(Ec * 32) / 256, 256, 0, stream>>>(
            m_m, xs_u, ld_u, ei_rates, ei_rates + Ec, Ec);
        count_kernel<<<(Ec + 255) / 256, 256, 0, stream>>>(cnt_m, ei_rates + Ec, Ec);
        scatter_add_kernel<<<(Ec * 32) / 256, 256, 0, stream>>>(
            m_u, xs_m, ld_m, ei_rev, ei_rev + Ec, Ec);
        count_kernel<<<(Ec + 255) / 256, 256, 0, stream>>>(cnt_u, ei_rev + Ec, Ec);

        normalize_kernel<<<(unsigned)(((size_t)NMc * Hc + 255) / 256), 256, 0, stream>>>(
            m_m, cnt_m, NMc);
        normalize_kernel<<<(unsigned)(((size_t)NUc * Hc + 255) / 256), 256, 0, stream>>>(
            m_u, cnt_u, NUc);

        // new_m = relu(m_m @ Wl_rates[l] + b_rates[l] + xm @ Wr_rates[l])
        wmma_gemm128<<<NMc / 16, 256, 0, stream>>>(
            xm_cat + l * Hc, 3 * Hc,
            m_m, Hc, Hc, xs_m, ld_m, Hc,
            Wl_rates + (size_t)l * Hc * Hc, Wr_rates + (size_t)l * Hc * Hc,
            b_rates + l * Hc, 1);
        // new_u = relu(m_u @ Wl_rev[l] + b_rev[l] + xu @ Wr_rev[l])
        wmma_gemm128<<<NUc / 16, 256, 0, stream>>>(
            xu_cat + l * Hc, 3 * Hc,
            m_u, Hc, Hc, xs_u, ld_u, Hc,
            Wl_rev + (size_t)l * Hc * Hc, Wr_rev + (size_t)l * Hc * Hc,
            b_rev + l * Hc, 1);

        xs_u = xu_cat + l * Hc; ld_u = 3 * Hc;
        xs_m = xm_cat + l * Hc; ld_m = 3 * Hc;
    }

    // 3) JumpingKnowledge cat + projection
    wmma_gemm128<<<NUc / 16, 256, 0, stream>>>(
        xu_proj, Hc, xu_cat, 3 * Hc, 3 * Hc, nullptr, 0, 0,
        Wp_user, nullptr, bp_user, 0);
    wmma_gemm128<<<NMc / 16, 256, 0, stream>>>(
        xm_proj, Hc, xm_cat, 3 * Hc, 3 * Hc, nullptr, 0, 0,
        Wp_movie, nullptr, bp_movie, 0);

    // 4) fused edge classifier
    classifier_kernel<<<ELc / 16, 256, 0, stream>>>(
        xu_proj, xm_proj, eli, eli + ELc,
        W1, b1, g1, be1, W2, b2, g2, be2, W3, b3, out);
}